// QANet_48928267436508
// MI455X (gfx1250) — compile-verified
//
#include <hip/hip_runtime.h>
#include <math.h>

typedef float v2f __attribute__((ext_vector_type(2)));
typedef float v8f __attribute__((ext_vector_type(8)));

// ---------------------------------------------------------------------------
// Generic batched strided GEMM on the CDNA5 f32 WMMA path.
//   C[m,n] = sum_k A[m,k]*B[k,n]  (+bias)(ReLU?)(+resid), fully strided.
// One wave (32 threads) computes a 32x16 tile: two 16x16 WMMA sub-tiles that
// share one B fragment. Inner loop is branch-free: OOB rows/cols are clamped
// (their results are never stored); the K remainder is handled by a peeled
// tail step that zeroes the B fragment rows with a 0/1 multiply.
// Batch offsets: off = ((bz / div) % mod) * stride for A, B, bias;
// C and resid are offset by bz * stride.
// ---------------------------------------------------------------------------
__global__ void k_gemm_wmma(const float* __restrict__ A,
                            const float* __restrict__ Bm,
                            float* __restrict__ C,
                            const float* __restrict__ bias,
                            const float* __restrict__ resid,
                            int M, int N, int Kd,
                            int sa_m, int sa_k, int divA, int modA, long strA,
                            int sb_k, int sb_n, int divB, int modB, long strB,
                            int sc_m, int sc_n, long strC, long strRes,
                            int bias_mode, int divBias, int modBias, long strBias,
                            int do_relu) {
  int bz = blockIdx.z;
  const float* Abase = A + (long)((bz / divA) % modA) * strA;
  const float* Bbase = Bm + (long)((bz / divB) % modB) * strB;
  float* Cb = C + (long)bz * strC;
  const float* Rb = resid ? resid + (long)bz * strRes : nullptr;
  const float* biasb = bias ? bias + (long)((bz / divBias) % modBias) * strBias
                            : nullptr;

  int tm = blockIdx.x * 32;        // 32 output rows per wave (2 sub-tiles)
  int tn = blockIdx.y * 16;
  int lane = threadIdx.x;          // 0..31 (wave32)
  int half = lane >> 4;            // 0: lanes 0-15, 1: lanes 16-31
  int lm = lane & 15;

  int row0 = tm + lm;              // A-fragment M index, sub-tile 0
  int row1 = tm + 16 + lm;         // sub-tile 1
  int col  = tn + lm;              // B-fragment N index
  // Clamp: OOB lanes load valid garbage that only feeds unstored outputs.
  int rA0 = min(row0, M - 1);
  int rA1 = min(row1, M - 1);
  int cB  = min(col,  N - 1);

  const float* pA0 = Abase + (long)rA0 * sa_m + (long)(half * 2) * sa_k;
  const float* pA1 = Abase + (long)rA1 * sa_m + (long)(half * 2) * sa_k;
  const float* pB  = Bbase + (long)(half * 2) * sb_k + (long)cB * sb_n;
  long stepA = 4L * sa_k;
  long stepB = 4L * sb_k;

  v8f acc0 = {0.f, 0.f, 0.f, 0.f, 0.f, 0.f, 0.f, 0.f};
  v8f acc1 = {0.f, 0.f, 0.f, 0.f, 0.f, 0.f, 0.f, 0.f};

  int kfull = Kd & ~3;             // multiple-of-4 part
#pragma unroll 2
  for (int k0 = 0; k0 < kfull; k0 += 4) {
    v2f a0, a1, b;
    a0.x = pA0[0];
    a0.y = pA0[sa_k];
    a1.x = pA1[0];
    a1.y = pA1[sa_k];
    b.x  = pB[0];
    b.y  = pB[sb_k];
    pA0 += stepA;
    pA1 += stepA;
    pB  += stepB;
    acc0 = __builtin_amdgcn_wmma_f32_16x16x4_f32(false, a0, false, b,
                                                 (short)0, acc0, false, false);
    acc1 = __builtin_amdgcn_wmma_f32_16x16x4_f32(false, a1, false, b,
                                                 (short)0, acc1, false, false);
  }
  if (kfull < Kd) {                // peeled K tail (Kd % 4 in {1,2,3})
    int ka  = kfull + half * 2;
    float m0 = (ka     < Kd) ? 1.f : 0.f;
    float m1 = (ka + 1 < Kd) ? 1.f : 0.f;
    int ka0 = min(ka,     Kd - 1);
    int ka1 = min(ka + 1, Kd - 1);
    v2f a0, a1, b;
    a0.x = Abase[(long)rA0 * sa_m + (long)ka0 * sa_k];
    a0.y = Abase[(long)rA0 * sa_m + (long)ka1 * sa_k];
    a1.x = Abase[(long)rA1 * sa_m + (long)ka0 * sa_k];
    a1.y = Abase[(long)rA1 * sa_m + (long)ka1 * sa_k];
    b.x  = Bbase[(long)ka0 * sb_k + (long)cB * sb_n] * m0;  // zero OOB K rows
    b.y  = Bbase[(long)ka1 * sb_k + (long)cB * sb_n] * m1;
    acc0 = __builtin_amdgcn_wmma_f32_16x16x4_f32(false, a0, false, b,
                                                 (short)0, acc0, false, false);
    acc1 = __builtin_amdgcn_wmma_f32_16x16x4_f32(false, a1, false, b,
                                                 (short)0, acc1, false, false);
  }

  int n = tn + lm;
#pragma unroll
  for (int r = 0; r < 8; ++r) {
    int mr = r + half * 8;         // C/D layout: VGPR r -> M=r / r+8
    int m0 = tm + mr;
    if (m0 < M && n < N) {
      float v = acc0[r];
      if (bias_mode == 1)      v += biasb[m0];
      else if (bias_mode == 2) v += biasb[n];
      if (do_relu) v = fmaxf(v, 0.f);
      long ci = (long)m0 * sc_m + (long)n * sc_n;
      if (Rb) v += Rb[ci];
      Cb[ci] = v;
    }
    int m1 = tm + 16 + mr;
    if (m1 < M && n < N) {
      float v = acc1[r];
      if (bias_mode == 1)      v += biasb[m1];
      else if (bias_mode == 2) v += biasb[n];
      if (do_relu) v = fmaxf(v, 0.f);
      long ci = (long)m1 * sc_m + (long)n * sc_n;
      if (Rb) v += Rb[ci];
      Cb[ci] = v;
    }
  }
}

// ---------------------------------------------------------------------------
// out[b,d,l] = x[b,d,l] + sin(l * 10000^(-(d - d%2)/D) + (d%2)*pi/2)
// ---------------------------------------------------------------------------
__global__ void k_add_pe(const float* __restrict__ x, float* __restrict__ out,
                         int Bn, int D, int L) {
  long total = (long)Bn * D * L;
  long idx = (long)blockIdx.x * blockDim.x + threadIdx.x;
  if (idx >= total) return;
  int l = (int)(idx % L);
  int d = (int)((idx / L) % D);
  int de = d & ~1;
  float freq = expf(-(float)de / (float)D * logf(10000.0f));
  float ang = (float)l * freq + ((d & 1) ? 1.5707963267948966f : 0.0f);
  out[idx] = x[idx] + sinf(ang);
}

// ---------------------------------------------------------------------------
// LayerNorm over channel dim D of [B,D,L]; torch-style: unbiased std (ddof=1),
// divide by (std + eps). One wave per (b,l) column.
// ---------------------------------------------------------------------------
__global__ void k_layernorm(const float* __restrict__ x,
                            const float* __restrict__ g,
                            const float* __restrict__ be,
                            float* __restrict__ out, int Bn, int D, int L) {
  int col = blockIdx.x * blockDim.y + threadIdx.y;
  if (col >= Bn * L) return;
  int b = col / L, l = col % L;
  const float* xp = x + (long)b * D * L + l;
  int lane = threadIdx.x;
  float s = 0.f, s2 = 0.f;
  for (int d = lane; d < D; d += 32) {
    float v = xp[(long)d * L];
    s += v;
    s2 += v * v;
  }
  for (int off = 16; off > 0; off >>= 1) {
    s  += __shfl_xor(s, off, 32);
    s2 += __shfl_xor(s2, off, 32);
  }
  float mu = s / (float)D;
  float var = (s2 - s * mu) / (float)(D - 1);
  float sd = sqrtf(fmaxf(var, 0.f));
  float inv = 1.0f / (sd + 1e-6f);
  float* op = out + (long)b * D * L + l;
  for (int d = lane; d < D; d += 32)
    op[(long)d * L] = g[d] * (xp[(long)d * L] - mu) * inv + be[d];
}

// ---------------------------------------------------------------------------
// Depthwise conv1d, kernel 7, pad 3: out[b,d,l] = bias[d] + sum_k w[d,k]*x[...]
// ---------------------------------------------------------------------------
__global__ void k_dwconv(const float* __restrict__ x,
                         const float* __restrict__ w,
                         const float* __restrict__ bias,
                         float* __restrict__ out, int Bn, int D, int L) {
  long total = (long)Bn * D * L;
  long idx = (long)blockIdx.x * blockDim.x + threadIdx.x;
  if (idx >= total) return;
  int l = (int)(idx % L);
  long bd = idx / L;
  int d = (int)(bd % D);
  const float* xp = x + bd * L;
  const float* wp = w + (long)d * 7;
  float acc = bias[d];
#pragma unroll
  for (int k = 0; k < 7; ++k) {
    int ll = l + k - 3;
    if (ll >= 0 && ll < L) acc += wp[k] * xp[ll];
  }
  out[idx] = acc;
}

// ---------------------------------------------------------------------------
// Fused multi-head attention core. q/k/v: [B*H, L, 16]. mask: [B, L] over keys.
// Output o: [B, L, H*16]. One wave per query row; K/V staged in LDS.
// ---------------------------------------------------------------------------
#define MAXJ 13  // ceil(400/32)
__global__ void k_attn(const float* __restrict__ q,
                       const float* __restrict__ kk_,
                       const float* __restrict__ vv_,
                       const float* __restrict__ mask,
                       float* __restrict__ o, int Bn, int Hh, int L) {
  extern __shared__ float lds[];
  int bh = blockIdx.x;
  int b = bh / Hh;
  int h = bh % Hh;
  float* Ks = lds;
  float* Vs = lds + (long)L * 16;
  const float* kp = kk_ + (long)bh * L * 16;
  const float* vp = vv_ + (long)bh * L * 16;
  int tid = threadIdx.y * 32 + threadIdx.x;
  int nthr = blockDim.y * 32;
  for (int t = tid; t < L * 16; t += nthr) {
    Ks[t] = kp[t];
    Vs[t] = vp[t];
  }
  __syncthreads();

  int i = blockIdx.y * blockDim.y + threadIdx.y;
  if (i >= L) return;
  int lane = threadIdx.x;
  const float* qp = q + ((long)bh * L + i) * 16;
  float qr[16];
#pragma unroll
  for (int c = 0; c < 16; ++c) qr[c] = qp[c];

  const float scale = 0.25f;  // 1/sqrt(16)
  float sc[MAXJ];
  float mx = -INFINITY;
  int cnt = 0;
  for (int j = lane; j < L; j += 32) {
    const float* kr = Ks + (long)j * 16;
    float s = 0.f;
#pragma unroll
    for (int c = 0; c < 16; ++c) s += qr[c] * kr[c];
    s = s * scale - 1e30f * (1.0f - mask[(long)b * L + j]);
    sc[cnt++] = s;
    mx = fmaxf(mx, s);
  }
  for (int off = 16; off > 0; off >>= 1) mx = fmaxf(mx, __shfl_xor(mx, off, 32));

  float ssum = 0.f;
  float acc[16];
#pragma unroll
  for (int c = 0; c < 16; ++c) acc[c] = 0.f;
  cnt = 0;
  for (int j = lane; j < L; j += 32) {
    float wgt = __expf(sc[cnt++] - mx);
    ssum += wgt;
    const float* vr = Vs + (long)j * 16;
#pragma unroll
    for (int c = 0; c < 16; ++c) acc[c] += wgt * vr[c];
  }
  for (int off = 16; off > 0; off >>= 1) {
    ssum += __shfl_xor(ssum, off, 32);
#pragma unroll
    for (int c = 0; c < 16; ++c) acc[c] += __shfl_xor(acc[c], off, 32);
  }
  float inv = 1.0f / ssum;
  if (lane == 0) {
    float* op = o + ((long)b * L + i) * (Hh * 16) + h * 16;
#pragma unroll
    for (int c = 0; c < 16; ++c) op[c] = acc[c] * inv;
  }
}

// out[b*L + l] = sum_d x[b,d,l] * w[d]   (x is [B,D,L]); one wave per column
__global__ void k_rowdot(const float* __restrict__ x,
                         const float* __restrict__ w,
                         float* __restrict__ out, int Bn, int D, int L) {
  int col = blockIdx.x * blockDim.y + threadIdx.y;
  if (col >= Bn * L) return;
  int b = col / L, l = col % L;
  const float* xp = x + (long)b * D * L + l;
  float s = 0.f;
  for (int d = threadIdx.x; d < D; d += 32) s += xp[(long)d * L] * w[d];
  for (int off = 16; off > 0; off >>= 1) s += __shfl_xor(s, off, 32);
  if (threadIdx.x == 0) out[col] = s;
}

// CW[b,n,d] = C[b,d,n] * wm[d]   ([B,D,N] -> [B,N,D] with per-d scale)
__global__ void k_scale_cw(const float* __restrict__ x,
                           const float* __restrict__ wm,
                           float* __restrict__ out, int Bn, int D, int L) {
  long total = (long)Bn * L * D;
  long idx = (long)blockIdx.x * blockDim.x + threadIdx.x;
  if (idx >= total) return;
  int d = (int)(idx % D);
  int n = (int)((idx / D) % L);
  int b = (int)(idx / ((long)D * L));
  out[idx] = x[(long)b * D * L + (long)d * L + n] * wm[d];
}

// S[b,n,m] += cdot[b,n] + qdot[b,m] + bias
__global__ void k_s_finalize(float* __restrict__ S,
                             const float* __restrict__ cdot,
                             const float* __restrict__ qdot,
                             const float* __restrict__ cqb,
                             int Bn, int Nn, int Mm) {
  long total = (long)Bn * Nn * Mm;
  long idx = (long)blockIdx.x * blockDim.x + threadIdx.x;
  if (idx >= total) return;
  int m = (int)(idx % Mm);
  int n = (int)((idx / Mm) % Nn);
  int b = (int)(idx / ((long)Mm * Nn));
  S[idx] += cdot[(long)b * Nn + n] + qdot[(long)b * Mm + m] + cqb[0];
}

// softmax over last dim Cc of [Bn, R, Cc]; mask [Bn, Cc] on that dim
__global__ void k_softmax_lastdim(const float* __restrict__ S,
                                  const float* __restrict__ mask,
                                  float* __restrict__ out,
                                  int Bn, int R, int Cc) {
  int row = blockIdx.x * blockDim.y + threadIdx.y;
  if (row >= Bn * R) return;
  int b = row / R;
  const float* sp = S + (long)row * Cc;
  float* op = out + (long)row * Cc;
  float mx = -INFINITY;
  for (int c = threadIdx.x; c < Cc; c += 32) {
    float v = sp[c] - 1e30f * (1.f - mask[(long)b * Cc + c]);
    mx = fmaxf(mx, v);
  }
  for (int off = 16; off > 0; off >>= 1) mx = fmaxf(mx, __shfl_xor(mx, off, 32));
  float sum = 0.f;
  for (int c = threadIdx.x; c < Cc; c += 32) {
    float v = sp[c] - 1e30f * (1.f - mask[(long)b * Cc + c]);
    sum += __expf(v - mx);
  }
  for (int off = 16; off > 0; off >>= 1) sum += __shfl_xor(sum, off, 32);
  float inv = 1.0f / sum;
  for (int c = threadIdx.x; c < Cc; c += 32) {
    float v = sp[c] - 1e30f * (1.f - mask[(long)b * Cc + c]);
    op[c] = __expf(v - mx) * inv;
  }
}

// softmax over middle dim R of [Bn, R, Cc]; mask [Bn, R] on that dim
__global__ void k_softmax_middim(const float* __restrict__ S,
                                 const float* __restrict__ mask,
                                 float* __restrict__ out,
                                 int Bn, int R, int Cc) {
  int col = blockIdx.x * blockDim.y + threadIdx.y;
  if (col >= Bn * Cc) return;
  int b = col / Cc, m = col % Cc;
  const float* sp = S + (long)b * R * Cc + m;
  float* op = out + (long)b * R * Cc + m;
  float mx = -INFINITY;
  for (int n = threadIdx.x; n < R; n += 32) {
    float v = sp[(long)n * Cc] - 1e30f * (1.f - mask[(long)b * R + n]);
    mx = fmaxf(mx, v);
  }
  for (int off = 16; off > 0; off >>= 1) mx = fmaxf(mx, __shfl_xor(mx, off, 32));
  float sum = 0.f;
  for (int n = threadIdx.x; n < R; n += 32) {
    float v = sp[(long)n * Cc] - 1e30f * (1.f - mask[(long)b * R + n]);
    sum += __expf(v - mx);
  }
  for (int off = 16; off > 0; off >>= 1) sum += __shfl_xor(sum, off, 32);
  float inv = 1.0f / sum;
  for (int n = threadIdx.x; n < R; n += 32) {
    float v = sp[(long)n * Cc] - 1e30f * (1.f - mask[(long)b * R + n]);
    op[(long)n * Cc] = __expf(v - mx) * inv;
  }
}

// out[b,n,:] = concat(C2, A, C2*A, C2*Bt);  C2[b,n,d] = curc[b,d,n]
__global__ void k_output(const float* __restrict__ curc,
                         const float* __restrict__ Ab,
                         const float* __restrict__ Bt,
                         float* __restrict__ out, int Bn, int D, int Nn) {
  long total = (long)Bn * Nn * 4 * D;
  long idx = (long)blockIdx.x * blockDim.x + threadIdx.x;
  if (idx >= total) return;
  int j = (int)(idx % (4 * D));
  int n = (int)((idx / (4 * D)) % Nn);
  int b = (int)(idx / ((long)4 * D * Nn));
  int d = j % D;
  int sec = j / D;
  float c2 = curc[(long)b * D * Nn + (long)d * Nn + n];
  long nd = ((long)b * Nn + n) * D + d;
  float v;
  if (sec == 0)      v = c2;
  else if (sec == 1) v = Ab[nd];
  else if (sec == 2) v = c2 * Ab[nd];
  else               v = c2 * Bt[nd];
  out[idx] = v;
}

// ---------------------------------------------------------------------------
extern "C" void kernel_launch(void* const* d_in, const int* in_sizes, int n_in,
                              void* d_out, int out_size, void* d_ws,
                              size_t ws_size, hipStream_t stream) {
  (void)in_sizes; (void)n_in; (void)out_size; (void)ws_size;
  const int B = 64, D = 128, N = 400, Mq = 50, H = 8, KD = 16;
  const int BIG = 1 << 30;

  const float* ctx    = (const float*)d_in[0];
  const float* que    = (const float*)d_in[1];
  const float* c_mask = (const float*)d_in[2];
  const float* q_mask = (const float*)d_in[3];
  const float* ln_g   = (const float*)d_in[4];
  const float* ln_b   = (const float*)d_in[5];
  const float* dw_w   = (const float*)d_in[6];
  const float* dw_b   = (const float*)d_in[7];
  const float* pw_w   = (const float*)d_in[8];
  const float* pw_b   = (const float*)d_in[9];
  const float* Wq = (const float*)d_in[10]; const float* bq = (const float*)d_in[11];
  const float* Wk = (const float*)d_in[12]; const float* bk = (const float*)d_in[13];
  const float* Wv = (const float*)d_in[14]; const float* bv = (const float*)d_in[15];
  const float* Wo = (const float*)d_in[16]; const float* bo = (const float*)d_in[17];
  const float* Wfc = (const float*)d_in[18]; const float* bfc = (const float*)d_in[19];
  const float* cq_wc = (const float*)d_in[20];
  const float* cq_wq = (const float*)d_in[21];
  const float* cq_wm = (const float*)d_in[22];
  const float* cq_b  = (const float*)d_in[23];
  float* out = (float*)d_out;

  size_t BDN = (size_t)B * D * N, BDM = (size_t)B * D * Mq;
  size_t BNM = (size_t)B * N * Mq;
  float* w = (float*)d_ws;
  float* curC = w;            float* lnC = w + BDN;      float* tmpC = w + 2 * BDN;
  float* qC = w + 3 * BDN;    float* kC = w + 4 * BDN;   float* vC = w + 5 * BDN;
  float* b2 = w + 6 * BDN;
  float* curQ = b2;           float* lnQ = b2 + BDM;     float* tmpQ = b2 + 2 * BDM;
  float* qQ = b2 + 3 * BDM;   float* kQ = b2 + 4 * BDM;  float* vQ = b2 + 5 * BDM;
  float* S    = b2 + 6 * BDM;
  float* Scb  = S + BNM;
  float* cdot = Scb + BNM;
  float* qdot = cdot + (size_t)B * N;
  float* U    = qdot + (size_t)B * Mq;   // B*Mq*D
  float* CW = qC;   // reuse context attention buffers (dead after encoders)
  float* Ab = kC;
  float* Bt = vC;

  auto gemm = [&](const float* A, const float* Bm, float* C, const float* bias,
                  const float* resid, int M, int Nn, int Kd,
                  int sa_m, int sa_k, int divA, int modA, long strA,
                  int sb_k, int sb_n, int divB, int modB, long strB,
                  int sc_m, int sc_n, long strC, long strRes,
                  int bias_mode, int divBias, int modBias, long strBias,
                  int relu, int batch) {
    dim3 g((M + 31) / 32, (Nn + 15) / 16, batch), blk(32, 1, 1);
    k_gemm_wmma<<<g, blk, 0, stream>>>(
        A, Bm, C, bias, resid, M, Nn, Kd, sa_m, sa_k, divA, modA, strA,
        sb_k, sb_n, divB, modB, strB, sc_m, sc_n, strC, strRes,
        bias_mode, divBias, modBias, strBias, relu);
  };

  auto encoder = [&](const float* x, const float* mask, int L, float* cur,
                     float* lnb, float* tmp, float* qb, float* kb, float* vb) {
    long total = (long)B * D * L;
    int tpb = 256;
    dim3 eg((unsigned)((total + tpb - 1) / tpb)), eb(tpb);
    k_add_pe<<<eg, eb, 0, stream>>>(x, cur, B, D, L);

    dim3 lnGrid((unsigned)((B * L + 7) / 8)), lnBlk(32, 8);
    for (int i = 0; i < 4; ++i) {
      k_layernorm<<<lnGrid, lnBlk, 0, stream>>>(cur, ln_g, ln_b, lnb, B, D, L);
      k_dwconv<<<eg, eb, 0, stream>>>(lnb, dw_w + (size_t)i * D * 7,
                                      dw_b + (size_t)i * D, tmp, B, D, L);
      // cur = relu(pw_w[i] @ tmp + pw_b[i]) + cur
      gemm(pw_w + (size_t)i * D * D, tmp, cur, pw_b + (size_t)i * D, cur,
           D, L, D,
           D, 1, 1, 1, 0L,
           L, 1, 1, BIG, (long)D * L,
           L, 1, (long)D * L, (long)D * L,
           1, 1, 1, 0L, 1, B);
    }
    // self-attention
    k_layernorm<<<lnGrid, lnBlk, 0, stream>>>(cur, ln_g, ln_b, lnb, B, D, L);
    const float* Ws[3] = {Wq, Wk, Wv};
    const float* bs[3] = {bq, bk, bv};
    float* os[3] = {qb, kb, vb};
    for (int t = 0; t < 3; ++t) {
      // [bz=b*H+h]: out[L,16] = lnb[b]^T[L,D] @ W[h][D,16] + bias[h]
      gemm(lnb, Ws[t], os[t], bs[t], nullptr,
           L, KD, D,
           1, L, H, BIG, (long)D * L,
           KD, 1, 1, H, (long)D * KD,
           KD, 1, (long)L * KD, 0L,
           2, 1, H, (long)KD, 0, B * H);
    }
    dim3 ag(B * H, (unsigned)((L + 7) / 8)), ab(32, 8);
    size_t shmem = (size_t)L * KD * 2 * sizeof(float);
    k_attn<<<ag, ab, shmem, stream>>>(qb, kb, vb, mask, tmp, B, H, L);
    // cur[b,d,l] = (tmp[b] @ Wo)[l,d] + bo[d] + cur[b,d,l]
    gemm(tmp, Wo, cur, bo, cur,
         L, D, H * KD,
         H * KD, 1, 1, BIG, (long)L * H * KD,
         D, 1, 1, 1, 0L,
         1, L, (long)D * L, (long)D * L,
         2, 1, 1, 0L, 0, B);
    // FFN: cur = Wfc @ LN(cur) + bfc + cur
    k_layernorm<<<lnGrid, lnBlk, 0, stream>>>(cur, ln_g, ln_b, lnb, B, D, L);
    gemm(Wfc, lnb, cur, bfc, cur,
         D, L, D,
         D, 1, 1, 1, 0L,
         L, 1, 1, BIG, (long)D * L,
         L, 1, (long)D * L, (long)D * L,
         1, 1, 1, 0L, 0, B);
  };

  encoder(ctx, c_mask, N, curC, lnC, tmpC, qC, kC, vC);
  encoder(que, q_mask, Mq, curQ, lnQ, tmpQ, qQ, kQ, vQ);

  // --------------------- context-query attention ---------------------------
  int tpb = 256;
  long totCW = (long)B * N * D;
  k_scale_cw<<<dim3((unsigned)((totCW + tpb - 1) / tpb)), dim3(tpb), 0, stream>>>(
      curC, cq_wm, CW, B, D, N);
  k_rowdot<<<dim3((unsigned)((B * N + 7) / 8)), dim3(32, 8), 0, stream>>>(
      curC, cq_wc, cdot, B, D, N);
  k_rowdot<<<dim3((unsigned)((B * Mq + 7) / 8)), dim3(32, 8), 0, stream>>>(
      curQ, cq_wq, qdot, B, D, Mq);
  // S0[b,n,m] = CW[b,n,:] . Q2[b,m,:],  Q2[b,m,d] = curQ[b,d,m]
  gemm(CW, curQ, S, nullptr, nullptr,
       N, Mq, D,
       D, 1, 1, BIG, (long)N * D,
       Mq, 1, 1, BIG, (long)D * Mq,
       Mq, 1, (long)N * Mq, 0L,
       0, 1, 1, 0L, 0, B);
  long totS = (long)B * N * Mq;
  k_s_finalize<<<dim3((unsigned)((totS + tpb - 1) / tpb)), dim3(tpb), 0, stream>>>(
      S, cdot, qdot, cq_b, B, N, Mq);
  // S_c (softmax over n) first — then S_r overwrites S in place
  k_softmax_middim<<<dim3((unsigned)((B * Mq + 7) / 8)), dim3(32, 8), 0, stream>>>(
      S, c_mask, Scb, B, N, Mq);
  k_softmax_lastdim<<<dim3((unsigned)((B * N + 7) / 8)), dim3(32, 8), 0, stream>>>(
      S, q_mask, S, B, N, Mq);
  // U[b,m,d] = sum_n Sc[b,n,m] * C2[b,n,d]
  gemm(Scb, curC, U, nullptr, nullptr,
       Mq, D, N,
       1, Mq, 1, BIG, (long)N * Mq,
       1, N, 1, BIG, (long)D * N,
       D, 1, (long)Mq * D, 0L,
       0, 1, 1, 0L, 0, B);
  // A[b,n,d] = S_r @ Q2
  gemm(S, curQ, Ab, nullptr, nullptr,
       N, D, Mq,
       Mq, 1, 1, BIG, (long)N * Mq,
       1, Mq, 1, BIG, (long)D * Mq,
       D, 1, (long)N * D, 0L,
       0, 1, 1, 0L, 0, B);
  // Bt[b,n,d] = S_r @ U   (== (S_r S_c^T) C2 by associativity)
  gemm(S, U, Bt, nullptr, nullptr,
       N, D, Mq,
       Mq, 1, 1, BIG, (long)N * Mq,
       D, 1, 1, BIG, (long)Mq * D,
       D, 1, (long)N * D, 0L,
       0, 1, 1, 0L, 0, B);

  long totO = (long)B * N * 4 * D;
  k_output<<<dim3((unsigned)((totO + tpb - 1) / tpb)), dim3(tpb), 0, stream>>>(
      curC, Ab, Bt, out, B, D, N);
}